// attentionFlow_51891794870346
// MI455X (gfx1250) — compile-verified
//
#include <hip/hip_runtime.h>
#include <math.h>
#include <stdint.h>

#define B_ 32
#define T_ 1024
#define J_ 128
#define H_ 1024
#define KC 64

typedef __attribute__((ext_vector_type(2))) float v2f;
typedef __attribute__((ext_vector_type(8))) float v8f;

__device__ __forceinline__ float wave_max16(float v) {
#pragma unroll
  for (int m = 1; m <= 8; m <<= 1) v = fmaxf(v, __shfl_xor(v, m, 32));
  return v;
}
__device__ __forceinline__ float wave_max32(float v) {
#pragma unroll
  for (int m = 1; m <= 16; m <<= 1) v = fmaxf(v, __shfl_xor(v, m, 32));
  return v;
}
__device__ __forceinline__ float wave_sum32(float v) {
#pragma unroll
  for (int m = 1; m <= 16; m <<= 1) v += __shfl_xor(v, m, 32);
  return v;
}

// ---------------- K0: s_h[b,t] = h . wh ; s_u[b,j] = u . wu ----------------
__global__ __launch_bounds__(256) void k0_rowdots(const float* __restrict__ h,
                                                  const float* __restrict__ u,
                                                  const float* __restrict__ w,
                                                  float* __restrict__ sh,
                                                  float* __restrict__ su) {
  const int wave = (blockIdx.x * blockDim.x + threadIdx.x) >> 5;
  const int lane = threadIdx.x & 31;
  const int nH = B_ * T_;
  const float* row;
  const float* wv;
  float* out;
  int oidx;
  if (wave < nH) {
    row = h + (size_t)wave * H_; wv = w; out = sh; oidx = wave;
  } else {
    const int g = wave - nH;
    if (g >= B_ * J_) return;
    row = u + (size_t)g * H_; wv = w + H_; out = su; oidx = g;
  }
  float acc = 0.f;
  for (int k = lane; k < H_; k += 32) acc += row[k] * wv[k];
  acc = wave_sum32(acc);
  if (lane == 0) out[oidx] = acc;
}

// ---------------- K1: S = (h*wm) @ u^T + s_h + s_u + bias; rowmax ----------
__global__ __launch_bounds__(128) void k1_scores(
    const float* __restrict__ h, const float* __restrict__ u,
    const float* __restrict__ w, const float* __restrict__ bias,
    const float* __restrict__ sh, const float* __restrict__ su,
    float* __restrict__ S, float* __restrict__ rowm) {
  __shared__ float lh[64][KC + 4];   // stride 68 -> conflict-free
  __shared__ float lu[J_][KC + 4];
  const int b = blockIdx.y;
  const int t0 = blockIdx.x * 64;
  const int tid = threadIdx.x;
  const int lane = tid & 31, wave = tid >> 5;
  const int ln = lane & 15, hi = lane >> 4;
  const int klo = hi * 2;
  const float* wm = w + 2 * H_;
  const float* ub = u + (size_t)(b * J_) * H_;   // uniform 64-bit base (SGPRs)
  const unsigned lu_base = (unsigned)(uintptr_t)&lu[0][0];  // LDS byte offset

  v8f acc[8] = {};

  for (int kc = 0; kc < H_; kc += KC) {
    // async-stage u tile (128 x 64) straight into LDS, no VGPR round-trip
#pragma unroll
    for (int i = 0; i < 16; ++i) {
      const int idx = tid + i * 128;           // 0..2047
      const int r = idx >> 4, c4 = (idx & 15) << 2;
      const unsigned ldsoff = lu_base + (unsigned)((r * (KC + 4) + c4) * 4);
      const unsigned goff = (unsigned)(((r * H_) + kc + c4) * 4);
      asm volatile("global_load_async_to_lds_b128 %0, %1, %2"
                   :: "v"(ldsoff), "v"(goff), "s"(ub)
                   : "memory");
    }
    // stage h tile (64 x 64), pre-scaled by wm (needs VALU, so stays manual)
#pragma unroll
    for (int i = 0; i < 8; ++i) {
      const int idx = tid + i * 128;           // 0..1023
      const int r = idx >> 4, c4 = (idx & 15) << 2;
      const float4 hv = *(const float4*)(h + ((size_t)(b * T_ + t0 + r) * H_ + kc + c4));
      const float4 wv = *(const float4*)(wm + kc + c4);
      lh[r][c4 + 0] = hv.x * wv.x;
      lh[r][c4 + 1] = hv.y * wv.y;
      lh[r][c4 + 2] = hv.z * wv.z;
      lh[r][c4 + 3] = hv.w * wv.w;
    }
    // prefetch next K-chunk of h into cache
    if (kc + KC < H_) {
      __builtin_prefetch(h + ((size_t)(b * T_ + t0 + (tid >> 1)) * H_ + kc + KC + (tid & 1) * 32), 0, 0);
    }
    asm volatile("s_wait_asynccnt 0x0" ::: "memory");
    __syncthreads();

    const float* lhp = &lh[wave * 16 + ln][0];
    for (int kk = 0; kk < KC; kk += 4) {
      v2f a;
      a.x = lhp[kk + klo];
      a.y = lhp[kk + klo + 1];
#pragma unroll
      for (int nt = 0; nt < 8; ++nt) {
        v2f bb;
        bb.x = lu[nt * 16 + ln][kk + klo];
        bb.y = lu[nt * 16 + ln][kk + klo + 1];
        acc[nt] = __builtin_amdgcn_wmma_f32_16x16x4_f32(false, a, false, bb,
                                                        (short)0, acc[nt], false, false);
      }
    }
    __syncthreads();
  }

  // epilogue: add s_h + s_u + bias, store S, row-max over j
  const float bval = bias[0];
  const int rbase = t0 + wave * 16 + hi * 8;
  float shv[8], rmax[8];
#pragma unroll
  for (int v = 0; v < 8; ++v) {
    shv[v] = sh[b * T_ + rbase + v];
    rmax[v] = -3.4e38f;
  }
#pragma unroll
  for (int nt = 0; nt < 8; ++nt) {
    const int col = nt * 16 + ln;
    const float suv = su[b * J_ + col];
#pragma unroll
    for (int v = 0; v < 8; ++v) {
      const float val = acc[nt][v] + shv[v] + suv + bval;
      S[(size_t)(b * T_ + rbase + v) * J_ + col] = val;
      rmax[v] = fmaxf(rmax[v], val);
    }
  }
#pragma unroll
  for (int v = 0; v < 8; ++v) rmax[v] = wave_max16(rmax[v]);
  if (ln == 0) {
#pragma unroll
    for (int v = 0; v < 8; ++v) rowm[b * T_ + rbase + v] = rmax[v];
  }
}

// ---------------- K2: per-(b,j) column max & 1/sum(exp) over T -------------
__global__ __launch_bounds__(256) void k2_colstats(const float* __restrict__ S,
                                                   float* __restrict__ cmax,
                                                   float* __restrict__ crcp) {
  const int gid = (blockIdx.x * blockDim.x + threadIdx.x) >> 5;
  const int lane = threadIdx.x & 31;
  const int b = gid >> 7;          // / J_
  const int j = gid & (J_ - 1);
  const float* col = S + (size_t)b * T_ * J_ + j;
  float mx = -3.4e38f;
  for (int t = lane; t < T_; t += 32) mx = fmaxf(mx, col[(size_t)t * J_]);
  mx = wave_max32(mx);
  float s = 0.f;
  for (int t = lane; t < T_; t += 32) s += __expf(col[(size_t)t * J_] - mx);
  s = wave_sum32(s);
  if (lane == 0) {
    cmax[gid] = mx;
    crcp[gid] = 1.0f / s;
  }
}

// ---------------- K3: b_t = softmax_T(rowmax) ------------------------------
__global__ __launch_bounds__(256) void k3_bt(const float* __restrict__ rowm,
                                             float* __restrict__ bt) {
  __shared__ float red[256];
  const int b = blockIdx.x, tid = threadIdx.x;
  float mv[4];
#pragma unroll
  for (int i = 0; i < 4; ++i) mv[i] = rowm[b * T_ + tid + i * 256];
  float mx = fmaxf(fmaxf(mv[0], mv[1]), fmaxf(mv[2], mv[3]));
  red[tid] = mx;
  __syncthreads();
  for (int s = 128; s > 0; s >>= 1) {
    if (tid < s) red[tid] = fmaxf(red[tid], red[tid + s]);
    __syncthreads();
  }
  mx = red[0];
  __syncthreads();
  float sum = 0.f;
#pragma unroll
  for (int i = 0; i < 4; ++i) sum += __expf(mv[i] - mx);
  red[tid] = sum;
  __syncthreads();
  for (int s = 128; s > 0; s >>= 1) {
    if (tid < s) red[tid] += red[tid + s];
    __syncthreads();
  }
  const float inv = 1.0f / red[0];
#pragma unroll
  for (int i = 0; i < 4; ++i) bt[b * T_ + tid + i * 256] = __expf(mv[i] - mx) * inv;
}

// ---------------- K4: q2c[b,d] = sum_t b_t * h -----------------------------
__global__ __launch_bounds__(256) void k4_q2c(const float* __restrict__ h,
                                              const float* __restrict__ bt,
                                              float* __restrict__ q2c) {
  const int b = blockIdx.y;
  const int d = blockIdx.x * blockDim.x + threadIdx.x;
  const float* hp = h + (size_t)(b * T_) * H_ + d;
  const float* bp = bt + b * T_;
  float acc = 0.f;
  for (int t = 0; t < T_; ++t) acc += bp[t] * hp[(size_t)t * H_];
  q2c[b * H_ + d] = acc;
}

// ---------------- K5: c2q = P @ u ; assemble G -----------------------------
__global__ __launch_bounds__(256) void k5_c2q_G(
    const float* __restrict__ h, const float* __restrict__ u,
    const float* __restrict__ S, const float* __restrict__ cmax,
    const float* __restrict__ crcp, const float* __restrict__ q2c,
    float* __restrict__ G) {
  __shared__ float lp[64][J_ + 4];   // stride 132 -> conflict-free
  const int b = blockIdx.z;
  const int t0 = blockIdx.x * 64;
  const int d0 = blockIdx.y * 128;
  const int tid = threadIdx.x;
  const int lane = tid & 31, wave = tid >> 5;
  const int ln = lane & 15, hi = lane >> 4;
  const int klo = hi * 2;
  const int mrow0 = (wave >> 1) * 16;
  const int ncol0 = (wave & 1) * 64;

  // build P tile (64 x 128) in LDS: exp(S - colmax) * colrcp
#pragma unroll
  for (int i = 0; i < 32; ++i) {
    const int idx = tid + i * 256;   // 0..8191
    const int r = idx >> 7, j = idx & 127;
    const float sval = S[(size_t)(b * T_ + t0 + r) * J_ + j];
    lp[r][j] = __expf(sval - cmax[b * J_ + j]) * crcp[b * J_ + j];
  }
  __syncthreads();

  v8f acc[4] = {};
  const float* lpp = &lp[mrow0 + ln][0];
  for (int kk = 0; kk < J_; kk += 4) {
    v2f a;
    a.x = lpp[kk + klo];
    a.y = lpp[kk + klo + 1];
#pragma unroll
    for (int nt = 0; nt < 4; ++nt) {
      const size_t ub = (size_t)(b * J_ + kk + klo) * H_ + d0 + ncol0 + nt * 16 + ln;
      v2f bb;
      bb.x = u[ub];
      bb.y = u[ub + H_];
      acc[nt] = __builtin_amdgcn_wmma_f32_16x16x4_f32(false, a, false, bb,
                                                      (short)0, acc[nt], false, false);
    }
  }

  // epilogue: write G = [h, c2q, h*c2q, h*q2c]
  const int rbase = t0 + mrow0 + hi * 8;
#pragma unroll
  for (int nt = 0; nt < 4; ++nt) {
    const int d = d0 + ncol0 + nt * 16 + ln;
    const float qv = q2c[b * H_ + d];
#pragma unroll
    for (int v = 0; v < 8; ++v) {
      const int t = rbase + v;
      const float c2qv = acc[nt][v];
      const float hv = h[(size_t)(b * T_ + t) * H_ + d];
      float* gp = G + (size_t)(b * T_ + t) * (4 * H_);
      gp[d] = hv;
      gp[H_ + d] = c2qv;
      gp[2 * H_ + d] = hv * c2qv;
      gp[3 * H_ + d] = hv * qv;
    }
  }
}

extern "C" void kernel_launch(void* const* d_in, const int* in_sizes, int n_in,
                              void* d_out, int out_size, void* d_ws, size_t ws_size,
                              hipStream_t stream) {
  const float* h = (const float*)d_in[0];
  const float* u = (const float*)d_in[1];
  const float* w = (const float*)d_in[2];
  const float* bias = (const float*)d_in[3];
  float* G = (float*)d_out;
  float* ws = (float*)d_ws;

  float* sh = ws;                                 // B*T
  float* su = sh + B_ * T_;                       // B*J
  float* S = su + B_ * J_;                        // B*T*J
  float* rowm = S + (size_t)B_ * T_ * J_;         // B*T
  float* bt = rowm + B_ * T_;                     // B*T
  float* cmax = bt + B_ * T_;                     // B*J
  float* crcp = cmax + B_ * J_;                   // B*J
  float* q2c = crcp + B_ * J_;                    // B*H

  k0_rowdots<<<dim3((B_ * T_ + B_ * J_) / 8), dim3(256), 0, stream>>>(h, u, w, sh, su);
  k1_scores<<<dim3(T_ / 64, B_), dim3(128), 0, stream>>>(h, u, w, bias, sh, su, S, rowm);
  k2_colstats<<<dim3(B_ * J_ / 8), dim3(256), 0, stream>>>(S, cmax, crcp);
  k3_bt<<<dim3(B_), dim3(256), 0, stream>>>(rowm, bt);
  k4_q2c<<<dim3(H_ / 256, B_), dim3(256), 0, stream>>>(h, bt, q2c);
  k5_c2q_G<<<dim3(T_ / 64, H_ / 128, B_), dim3(256), 0, stream>>>(h, u, S, cmax, crcp, q2c, G);
}